// LossFunc_87179246174895
// MI455X (gfx1250) — compile-verified
//
#include <hip/hip_runtime.h>
#include <stdint.h>

typedef unsigned int u32;
typedef float v2f __attribute__((ext_vector_type(2)));
typedef float v8f __attribute__((ext_vector_type(8)));

#if defined(__has_builtin)
#if __has_builtin(__builtin_amdgcn_wmma_f32_16x16x4_f32)
#define HAVE_WMMA_F32X4 1
#endif
#endif

#define NIMG 16
#define HW 409600
#define HW4 102400
#define NTHREADS 256
#define CNT_BLOCKS 64
#define FIN_BLOCKS 100
#define NPART (NIMG * FIN_BLOCKS)

// ---- workspace layout (u32 units) ----
#define OFF_POS   0
#define OFF_NEG   16
#define OFF_HA    32
#define OFF_HN    (OFF_HA + NIMG * 2048)
#define OFF_PREFA (OFF_HN + NIMG * 2048)
#define OFF_KA    (OFF_PREFA + 16)
#define OFF_ACTA  (OFF_KA + 16)
#define OFF_PREFN (OFF_ACTA + 16)
#define OFF_KN    (OFF_PREFN + 16)
#define OFF_ACTN  (OFF_KN + 16)
#define OFF_THRA  (OFF_ACTN + 16)   /* float */
#define OFF_THRN  (OFF_THRA + 16)   /* float */
#define OFF_BR    (OFF_THRN + 16)
#define OFF_PART  (OFF_BR + 16)     /* float[4][NPART] */

#define C4(v, j) (((const float*)&(v))[j])

// order-preserving float->uint key (descending == larger key)
__device__ __forceinline__ u32 keyOf(float f) {
  u32 b = __float_as_uint(f);
  return b ^ ((u32)((int)b >> 31) | 0x80000000u);
}
__device__ __forceinline__ float unkey(u32 k) {
  u32 b = (k & 0x80000000u) ? (k ^ 0x80000000u) : ~k;
  return __uint_as_float(b);
}

__global__ void zero_u32(u32* __restrict__ p, int n) {
  int i = blockIdx.x * blockDim.x + threadIdx.x;
  if (i < n) p[i] = 0u;
}

// Pass 0: per-image pos/neg counts + top-11-bit histograms (all scores & negative scores)
__global__ void count_hist0(const float4* __restrict__ sc, const float4* __restrict__ gt,
                            const float4* __restrict__ tm, u32* __restrict__ ws) {
  __shared__ u32 hA[2048], hN[2048];
  __shared__ u32 rp[NTHREADS], rn[NTHREADS];
  const int tid = threadIdx.x, img = blockIdx.y;
  for (int i = tid; i < 2048; i += NTHREADS) { hA[i] = 0u; hN[i] = 0u; }
  __syncthreads();
  const float4* s4 = sc + (size_t)img * HW4;
  const float4* g4 = gt + (size_t)img * HW4;
  const float4* t4 = tm + (size_t)img * HW4;
  u32 pos = 0, neg = 0;
  for (int i = blockIdx.x * NTHREADS + tid; i < HW4; i += CNT_BLOCKS * NTHREADS) {
    float4 s = s4[i], g = g4[i], t = t4[i];
#pragma unroll
    for (int j = 0; j < 4; ++j) {
      float sv = C4(s, j), gv = C4(g, j), tv = C4(t, j);
      u32 k = keyOf(sv);
      atomicAdd(&hA[k >> 21], 1u);
      if (gv < 0.5f) { ++neg; atomicAdd(&hN[k >> 21], 1u); }
      if (gv > 0.5f && tv > 0.5f) ++pos;
    }
  }
  __syncthreads();
  u32* gA = ws + OFF_HA + img * 2048;
  u32* gN = ws + OFF_HN + img * 2048;
  for (int i = tid; i < 2048; i += NTHREADS) {
    if (hA[i]) atomicAdd(&gA[i], hA[i]);
    if (hN[i]) atomicAdd(&gN[i], hN[i]);
  }
  rp[tid] = pos; rn[tid] = neg;
  __syncthreads();
  for (int s2 = NTHREADS / 2; s2 > 0; s2 >>= 1) {
    if (tid < s2) { rp[tid] += rp[tid + s2]; rn[tid] += rn[tid + s2]; }
    __syncthreads();
  }
  if (tid == 0) {
    atomicAdd(&ws[OFF_POS + img], rp[0]);
    atomicAdd(&ws[OFF_NEG + img], rn[0]);
  }
}

// Radix-select refinement histograms (pass 1: bits 20..10, pass 2: bits 9..0)
__global__ void hist_pass(const float4* __restrict__ sc, const float4* __restrict__ gt,
                          u32* __restrict__ ws, int pass) {
  __shared__ u32 hA[2048], hN[2048];
  const int tid = threadIdx.x, img = blockIdx.y;
  for (int i = tid; i < 2048; i += NTHREADS) { hA[i] = 0u; hN[i] = 0u; }
  __syncthreads();
  const u32 prefA = ws[OFF_PREFA + img];
  const u32 prefN = ws[OFF_PREFN + img];
  const u32 actN  = ws[OFF_ACTN + img];
  const float4* s4 = sc + (size_t)img * HW4;
  const float4* g4 = gt + (size_t)img * HW4;
  for (int i = blockIdx.x * NTHREADS + tid; i < HW4; i += CNT_BLOCKS * NTHREADS) {
    float4 s = s4[i], g = g4[i];
#pragma unroll
    for (int j = 0; j < 4; ++j) {
      u32 k = keyOf(C4(s, j));
      u32 bin; bool mA, mN;
      if (pass == 1) {
        mA = (k >> 21) == (prefA >> 21);
        mN = (k >> 21) == (prefN >> 21);
        bin = (k >> 10) & 2047u;
      } else {
        mA = (k >> 10) == (prefA >> 10);
        mN = (k >> 10) == (prefN >> 10);
        bin = k & 1023u;
      }
      if (mA) atomicAdd(&hA[bin], 1u);
      if (actN && mN && C4(g, j) < 0.5f) atomicAdd(&hN[bin], 1u);
    }
  }
  __syncthreads();
  u32* gA = ws + OFF_HA + img * 2048;
  u32* gN = ws + OFF_HN + img * 2048;
  for (int i = tid; i < 2048; i += NTHREADS) {
    if (hA[i]) atomicAdd(&gA[i], hA[i]);
    if (hN[i]) atomicAdd(&gN[i], hN[i]);
  }
}

// One block per (image, selection): scan bins top-down, refine prefix and rank.
__global__ void select_pass(u32* __restrict__ ws, int pass) {
  __shared__ u32 h[2048];
  const int tid = threadIdx.x;
  const int sel = blockIdx.x & 1;   // 0 = all scores, 1 = negative scores
  const int img = blockIdx.x >> 1;
  const u32* hist = ws + (sel ? OFF_HN : OFF_HA) + img * 2048;
  for (int i = tid; i < 2048; i += NTHREADS) h[i] = hist[i];
  __syncthreads();
  if (tid != 0) return;
  u32 pref, k, active;
  if (pass == 0) {
    u32 pos = ws[OFF_POS + img], negc = ws[OFF_NEG + img];
    if (sel == 0) { u32 n1 = negc >> 1; k = n1 ? n1 : 1u; active = 1u; }
    else {
      unsigned long long n3 = 3ull * (unsigned long long)pos;
      if (n3 > negc) n3 = negc;
      k = (u32)n3; active = (k >= 1u) ? 1u : 0u;
    }
    pref = 0u;
  } else {
    pref   = ws[(sel ? OFF_PREFN : OFF_PREFA) + img];
    k      = ws[(sel ? OFF_KN    : OFF_KA)    + img];
    active = ws[(sel ? OFF_ACTN  : OFF_ACTA)  + img];
  }
  if (active) {
    const int nb = (pass == 2) ? 1024 : 2048;
    u32 cum = 0; int bsel = 0; u32 krem = k;
    for (int b = nb - 1; b >= 0; --b) {
      u32 c = h[b];
      if (cum + c >= k) { bsel = b; krem = k - cum; break; }
      cum += c;
    }
    k = krem;
    if (pass == 0)      pref = ((u32)bsel) << 21;
    else if (pass == 1) pref |= ((u32)bsel) << 10;
    else                pref |= (u32)bsel;
  }
  ws[(sel ? OFF_PREFN : OFF_PREFA) + img] = pref;
  ws[(sel ? OFF_KN    : OFF_KA)    + img] = k;
  ws[(sel ? OFF_ACTN  : OFF_ACTA)  + img] = active;
  if (pass == 2) {
    ((float*)ws)[(sel ? OFF_THRN : OFF_THRA) + img] = active ? unkey(pref) : 0.0f;
    if (sel == 1) {
      u32 pos = ws[OFF_POS + img], negc = ws[OFF_NEG + img];
      unsigned long long n3 = 3ull * (unsigned long long)pos;
      if (n3 > negc) n3 = negc;
      ws[OFF_BR + img] = (pos == 0u) ? 0u : ((n3 == 0ull) ? 1u : 2u);
    }
  }
}

// Fused: OHEM mask + dice partials + geometry loss partials
__global__ void final_sums(const float4* __restrict__ ytc, const float4* __restrict__ ypc,
                           const float4* __restrict__ ytg, const float4* __restrict__ ypg,
                           const float4* __restrict__ tmk, const u32* __restrict__ ws,
                           float* __restrict__ part) {
  const int tid = threadIdx.x, img = blockIdx.y;
  const u32 br = ws[OFF_BR + img];
  const float thr1 = ((const float*)ws)[OFF_THRA + img];
  const float thr3 = ((const float*)ws)[OFF_THRN + img];
  const float4* g4 = ytc + (size_t)img * HW4;
  const float4* s4 = ypc + (size_t)img * HW4;
  const float4* t4 = tmk + (size_t)img * HW4;
  const float4* tg = ytg + (size_t)img * 5 * HW4;
  const float4* pg = ypg + (size_t)img * 5 * HW4;
  float aI = 0.f, aT = 0.f, aP = 0.f, aG = 0.f;
  for (int i = blockIdx.x * NTHREADS + tid; i < HW4; i += FIN_BLOCKS * NTHREADS) {
    float4 g = g4[i], s = s4[i], t = t4[i];
    float4 d1g = tg[i],           d2g = tg[i + HW4],     d3g = tg[i + 2 * HW4];
    float4 d4g = tg[i + 3 * HW4], thg = tg[i + 4 * HW4];
    float4 d1p = pg[i],           d2p = pg[i + HW4],     d3p = pg[i + 2 * HW4];
    float4 d4p = pg[i + 3 * HW4], thp = pg[i + 4 * HW4];
#pragma unroll
    for (int j = 0; j < 4; ++j) {
      float gg = C4(g, j), ss = C4(s, j), tt = C4(t, j);
      float m;
      if (br == 0u)      m = (ss >= thr1) ? 1.f : 0.f;
      else if (br == 1u) m = tt;
      else               m = (((ss >= thr3) || (gg > 0.5f)) && (tt > 0.5f)) ? 1.f : 0.f;
      aI += gg * ss * m; aT += gg * m; aP += ss * m;
      float x1g = C4(d1g, j), x2g = C4(d2g, j), x3g = C4(d3g, j), x4g = C4(d4g, j);
      float x1p = C4(d1p, j), x2p = C4(d2p, j), x3p = C4(d3p, j), x4p = C4(d4p, j);
      float a_gt = (x1g + x3g) * (x2g + x4g);
      float a_pr = (x1p + x3p) * (x2p + x4p);
      float wu = fminf(x2g, x2p) + fminf(x4g, x4p);
      float hu = fminf(x1g, x1p) + fminf(x3g, x3p);
      float ai = wu * hu;
      float au = a_gt + a_pr - ai;
      float laabb = -__logf((ai + 1.f) / (au + 1.f));
      float lth = 1.f - __cosf(C4(thp, j) - C4(thg, j));
      aG += (laabb + 20.f * lth) * gg * tt;
    }
  }
  __shared__ float r0[NTHREADS], r1[NTHREADS], r2[NTHREADS], r3[NTHREADS];
  r0[tid] = aI; r1[tid] = aT; r2[tid] = aP; r3[tid] = aG;
  __syncthreads();
  for (int s2 = NTHREADS / 2; s2 > 0; s2 >>= 1) {
    if (tid < s2) {
      r0[tid] += r0[tid + s2]; r1[tid] += r1[tid + s2];
      r2[tid] += r2[tid + s2]; r3[tid] += r3[tid + s2];
    }
    __syncthreads();
  }
  if (tid == 0) {
    int bl = img * FIN_BLOCKS + blockIdx.x;
    part[bl]             = r0[0];
    part[NPART + bl]     = r1[0];
    part[2 * NPART + bl] = r2[0];
    part[3 * NPART + bl] = r3[0];
  }
}

// Intra-wave sum of 32 lane values via V_WMMA_F32_16X16X4_F32 with all-ones A:
// D[m][n] = sum_k B[k][n]; summing row 0 over its 16 columns yields the full sum
// of all 64 B entries (32 data + 32 zeros) — layout-permutation invariant.
__device__ __forceinline__ float wave_reduce(float s) {
#ifdef HAVE_WMMA_F32X4
  v2f a; a[0] = 1.0f; a[1] = 1.0f;
  v2f b; b[0] = s;    b[1] = 0.0f;
  v8f c = {};
  c = __builtin_amdgcn_wmma_f32_16x16x4_f32(false, a, false, b, (short)0, c, false, false);
  float v = c[0];
  v += __shfl_xor(v, 1, 16);
  v += __shfl_xor(v, 2, 16);
  v += __shfl_xor(v, 4, 16);
  v += __shfl_xor(v, 8, 16);
  return v;
#else
  for (int off = 16; off >= 1; off >>= 1) s += __shfl_xor(s, off, 32);
  return s;
#endif
}

__global__ void finalize(const float* __restrict__ part, float* __restrict__ out) {
  const int tid = threadIdx.x;
  float s0 = 0.f, s1 = 0.f, s2 = 0.f, s3 = 0.f;
  for (int i = tid; i < NPART; i += NTHREADS) {
    s0 += part[i];
    s1 += part[NPART + i];
    s2 += part[2 * NPART + i];
    s3 += part[3 * NPART + i];
  }
  float r0 = wave_reduce(s0), r1 = wave_reduce(s1);
  float r2 = wave_reduce(s2), r3 = wave_reduce(s3);
  __shared__ float wsum[8][4];
  int wave = tid >> 5, lane = tid & 31;
  if (lane == 0) { wsum[wave][0] = r0; wsum[wave][1] = r1; wsum[wave][2] = r2; wsum[wave][3] = r3; }
  __syncthreads();
  if (tid == 0) {
    float inter = 0.f, st = 0.f, sp = 0.f, sg = 0.f;
    for (int w = 0; w < 8; ++w) {
      inter += wsum[w][0]; st += wsum[w][1]; sp += wsum[w][2]; sg += wsum[w][3];
    }
    float dice = 1.f - 2.f * inter / (st + sp + 1e-5f);
    out[0] = sg / (float)(NIMG * HW) + 0.01f * dice;
  }
}

extern "C" void kernel_launch(void* const* d_in, const int* in_sizes, int n_in,
                              void* d_out, int out_size, void* d_ws, size_t ws_size,
                              hipStream_t stream) {
  (void)in_sizes; (void)n_in; (void)out_size; (void)ws_size;
  const float4* ytc = (const float4*)d_in[0];
  const float4* ypc = (const float4*)d_in[1];
  const float4* ytg = (const float4*)d_in[2];
  const float4* ypg = (const float4*)d_in[3];
  const float4* tmk = (const float4*)d_in[4];
  u32* ws = (u32*)d_ws;
  float* part = (float*)(ws + OFF_PART);

  const int nzero0 = 32 + 2 * NIMG * 2048;              // counts + both hist banks
  const int nzeroH = 2 * NIMG * 2048;
  zero_u32<<<(nzero0 + 255) / 256, 256, 0, stream>>>(ws, nzero0);
  count_hist0<<<dim3(CNT_BLOCKS, NIMG), NTHREADS, 0, stream>>>(ypc, ytc, tmk, ws);
  select_pass<<<2 * NIMG, NTHREADS, 0, stream>>>(ws, 0);
  zero_u32<<<(nzeroH + 255) / 256, 256, 0, stream>>>(ws + OFF_HA, nzeroH);
  hist_pass<<<dim3(CNT_BLOCKS, NIMG), NTHREADS, 0, stream>>>(ypc, ytc, ws, 1);
  select_pass<<<2 * NIMG, NTHREADS, 0, stream>>>(ws, 1);
  zero_u32<<<(nzeroH + 255) / 256, 256, 0, stream>>>(ws + OFF_HA, nzeroH);
  hist_pass<<<dim3(CNT_BLOCKS, NIMG), NTHREADS, 0, stream>>>(ypc, ytc, ws, 2);
  select_pass<<<2 * NIMG, NTHREADS, 0, stream>>>(ws, 2);
  final_sums<<<dim3(FIN_BLOCKS, NIMG), NTHREADS, 0, stream>>>(ytc, ypc, ytg, ypg, tmk, ws, part);
  finalize<<<1, NTHREADS, 0, stream>>>(part, (float*)d_out);
}